// FPModule_8761733284509
// MI455X (gfx1250) — compile-verified
//
#include <hip/hip_runtime.h>
#include <hip/hip_bf16.h>

typedef __attribute__((ext_vector_type(2))) float v2f;
typedef __attribute__((ext_vector_type(8))) float v8f;

#define B_    4
#define N_    8192
#define S_    2048
#define D1_   128
#define D2_   256
#define CIN_  384
#define CMID_ 256
#define COUT_ 256
#define BN_   (B_ * N_)   // 32768 columns in the fused GEMMs

// GEMM tiling. LDS tiles are K-pair interleaved: element (k, col) lives at
// [k/2][col*2 + (k&1)], so a WMMA fragment (two consecutive K at one col) is
// a single aligned ds_load_b64 straight into an even VGPR pair.
#define TM_    64
#define TN_    128
#define TK_    32
#define NPAIR_ (TK_ / 2)
#define ASTR_  (TM_ * 2 + 32)   // 160; stride%64==32 -> halves on disjoint banks
#define BSTR_  (TN_ * 2 + 32)   // 288; stride%64==32 -> halves on disjoint banks

// ---------------------------------------------------------------------------
// Kernel 1: brute-force 3-NN + inverse-distance weights.
// pos2 tile (S*3 floats = 24 KB) staged in LDS; each thread owns one query.
// ---------------------------------------------------------------------------
__global__ void __launch_bounds__(256)
three_nn_kernel(const float* __restrict__ pos1, const float* __restrict__ pos2,
                int* __restrict__ nnIdx, float* __restrict__ nnW) {
    __shared__ float sp[S_ * 3];
    const int b = blockIdx.y;
    for (int t = threadIdx.x; t < 3 * S_; t += blockDim.x) {
        const int c = t / S_, s = t % S_;
        sp[s * 3 + c] = pos2[(b * 3 + c) * S_ + s];
    }
    __syncthreads();

    const int n = blockIdx.x * blockDim.x + threadIdx.x;
    const float px = pos1[(b * 3 + 0) * N_ + n];
    const float py = pos1[(b * 3 + 1) * N_ + n];
    const float pz = pos1[(b * 3 + 2) * N_ + n];

    float d0 = 3.4e38f, d1 = 3.4e38f, d2 = 3.4e38f;
    int   i0 = 0, i1 = 0, i2 = 0;
    for (int s = 0; s < S_; ++s) {
        const float dx = sp[3 * s + 0] - px;
        const float dy = sp[3 * s + 1] - py;
        const float dz = sp[3 * s + 2] - pz;
        const float d  = dx * dx + dy * dy + dz * dz;
        if (d < d0)      { d2 = d1; i2 = i1; d1 = d0; i1 = i0; d0 = d; i0 = s; }
        else if (d < d1) { d2 = d1; i2 = i1; d1 = d;  i1 = s; }
        else if (d < d2) { d2 = d;  i2 = s; }
    }
    d0 = fmaxf(d0, 1e-10f); d1 = fmaxf(d1, 1e-10f); d2 = fmaxf(d2, 1e-10f);
    const float w0 = 1.f / d0, w1 = 1.f / d1, w2 = 1.f / d2;
    const float rs = 1.f / (w0 + w1 + w2);
    const int j = b * N_ + n;
    nnIdx[j * 3 + 0] = i0; nnIdx[j * 3 + 1] = i1; nnIdx[j * 3 + 2] = i2;
    nnW[j * 3 + 0] = w0 * rs; nnW[j * 3 + 1] = w1 * rs; nnW[j * 3 + 2] = w2 * rs;
}

// ---------------------------------------------------------------------------
// Kernel 2: interpolated-gather + concat into feat[C_in][B*N] (GEMM layout).
// ---------------------------------------------------------------------------
__global__ void __launch_bounds__(256)
interp_concat_kernel(const float* __restrict__ feature1,
                     const float* __restrict__ feature2,
                     const int* __restrict__ nnIdx, const float* __restrict__ nnW,
                     float* __restrict__ feat) {
    const int c = blockIdx.y;
    const int j = blockIdx.x * blockDim.x + threadIdx.x;  // j = b*N + n
    const int b = j / N_;
    const int n = j - b * N_;
    float v;
    if (c < D2_) {
        const int*   ip = nnIdx + j * 3;
        const float* wp = nnW + j * 3;
        const float* f2 = feature2 + (size_t)(b * D2_ + c) * S_;
        v = wp[0] * f2[ip[0]] + wp[1] * f2[ip[1]] + wp[2] * f2[ip[2]];
    } else {
        v = feature1[(size_t)(b * D1_ + (c - D2_)) * N_ + n];
    }
    feat[(size_t)c * BN_ + j] = v;
}

// ---------------------------------------------------------------------------
// Kernel 3: LDS-tiled, double-buffered f32 WMMA GEMM.
//   Y[M x NC] = W[M x K] * X[K x NC] + bias
// Block: 256 threads (8 waves) -> 64(M) x 128(N) tile, K chunks of 32.
// Wave (mStrip = wave&3, nHalf = wave>>2) owns rows [mStrip*16,+16) x
// cols [nHalf*64,+64) as four v_wmma_f32_16x16x4_f32 accumulators.
// K-pair-interleaved LDS tiles -> every fragment is one ds_load_b64.
// Next chunk's global loads are issued before compute (reg staging), written
// to the alternate LDS buffer after compute -> global latency overlaps WMMA.
// ---------------------------------------------------------------------------
__global__ void __launch_bounds__(256)
gemm_f32_wmma_kernel(const float* __restrict__ Wm, const float* __restrict__ bias,
                     const float* __restrict__ X, float* __restrict__ Y,
                     int M, int K, int NC) {
    __shared__ float As[2][NPAIR_ * ASTR_];   // 2 x 2560 floats
    __shared__ float Bs[2][NPAIR_ * BSTR_];   // 2 x 4608 floats

    const int tid  = threadIdx.x;
    const int lane = tid & 31;
    const int wave = tid >> 5;
    const int half = lane >> 4;           // selects K-pair row offset
    const int lr   = lane & 15;           // row (A) / col (B,D) within 16
    const int mStrip = wave & 3;
    const int nHalf  = wave >> 2;

    const int rowBase0 = blockIdx.y * TM_;            // block's first M row
    const int tileN    = blockIdx.x * TN_;            // block's first column
    const int mB = mStrip * 16;                       // wave's M offset in tile
    const int nB = nHalf * 64;                        // wave's N offset in tile

    // cooperative-load mappings
    const int am = tid >> 2;              // 0..63  : A row within tile
    const int ap = (tid & 3) * 4;         // 0,4,8,12: first A K-pair (4 pairs/thread)
    const int bp = tid >> 4;              // 0..15  : B K-pair (rows 2bp, 2bp+1)
    const int bc = (tid & 15) * 8;        // 0..120 : B col offset (8 cols/thread)

    const int nK = K / TK_;

    // ---- prologue: stage chunk 0 into buffer 0 ----
    {
        const float* wrow = Wm + (size_t)(rowBase0 + am) * K + ap * 2;
        float4 a0 = *(const float4*)(wrow + 0);
        float4 a1 = *(const float4*)(wrow + 4);
        const float* xe = X + (size_t)(2 * bp) * NC + tileN + bc;      // even K row
        const float* xo = xe + NC;                                     // odd  K row
        float4 e0 = *(const float4*)(xe + 0);
        float4 e1 = *(const float4*)(xe + 4);
        float4 o0 = *(const float4*)(xo + 0);
        float4 o1 = *(const float4*)(xo + 4);
        float* arow = &As[0][0];
        *(v2f*)&arow[(ap + 0) * ASTR_ + am * 2] = v2f{a0.x, a0.y};
        *(v2f*)&arow[(ap + 1) * ASTR_ + am * 2] = v2f{a0.z, a0.w};
        *(v2f*)&arow[(ap + 2) * ASTR_ + am * 2] = v2f{a1.x, a1.y};
        *(v2f*)&arow[(ap + 3) * ASTR_ + am * 2] = v2f{a1.z, a1.w};
        float* brow = &Bs[0][bp * BSTR_ + bc * 2];
        *(v2f*)&brow[0]  = v2f{e0.x, o0.x}; *(v2f*)&brow[2]  = v2f{e0.y, o0.y};
        *(v2f*)&brow[4]  = v2f{e0.z, o0.z}; *(v2f*)&brow[6]  = v2f{e0.w, o0.w};
        *(v2f*)&brow[8]  = v2f{e1.x, o1.x}; *(v2f*)&brow[10] = v2f{e1.y, o1.y};
        *(v2f*)&brow[12] = v2f{e1.z, o1.z}; *(v2f*)&brow[14] = v2f{e1.w, o1.w};
    }
    __syncthreads();

    v8f acc[4] = {};

    for (int kt = 0; kt < nK; ++kt) {
        const int cur = kt & 1;
        const int nxt = cur ^ 1;
        const bool more = (kt + 1) < nK;

        // ---- issue next chunk's global loads early (overlap with WMMA) ----
        float4 a0, a1, e0, e1, o0, o1;
        if (more) {
            const int k0 = (kt + 1) * TK_;
            const float* wrow = Wm + (size_t)(rowBase0 + am) * K + k0 + ap * 2;
            a0 = *(const float4*)(wrow + 0);
            a1 = *(const float4*)(wrow + 4);
            const float* xe = X + (size_t)(k0 + 2 * bp) * NC + tileN + bc;
            const float* xo = xe + NC;
            e0 = *(const float4*)(xe + 0);
            e1 = *(const float4*)(xe + 4);
            o0 = *(const float4*)(xo + 0);
            o1 = *(const float4*)(xo + 4);
            if (kt + 2 < nK) {  // pull chunk-after-next of X toward L2/L0
                __builtin_prefetch(X + (size_t)((kt + 2) * TK_ + 2 * bp) * NC + tileN + bc, 0, 1);
            }
        }

        // ---- compute from current LDS buffer: 8 k-steps x 4 WMMAs ----
        const float* Ac = As[cur];
        const float* Bc = Bs[cur];
#pragma unroll
        for (int kk = 0; kk < TK_; kk += 4) {
            const int p = (kk >> 1) + half;          // kA = kk + 2*half (even)
            const v2f a = *(const v2f*)&Ac[p * ASTR_ + (mB + lr) * 2];
#pragma unroll
            for (int jt = 0; jt < 4; ++jt) {
                const v2f bf = *(const v2f*)&Bc[p * BSTR_ + (nB + jt * 16 + lr) * 2];
                acc[jt] = __builtin_amdgcn_wmma_f32_16x16x4_f32(
                    false, a, false, bf, (short)0, acc[jt], false, false);
            }
        }

        // ---- commit staged registers to the alternate buffer ----
        if (more) {
            float* arow = &As[nxt][0];
            *(v2f*)&arow[(ap + 0) * ASTR_ + am * 2] = v2f{a0.x, a0.y};
            *(v2f*)&arow[(ap + 1) * ASTR_ + am * 2] = v2f{a0.z, a0.w};
            *(v2f*)&arow[(ap + 2) * ASTR_ + am * 2] = v2f{a1.x, a1.y};
            *(v2f*)&arow[(ap + 3) * ASTR_ + am * 2] = v2f{a1.z, a1.w};
            float* brow = &Bs[nxt][bp * BSTR_ + bc * 2];
            *(v2f*)&brow[0]  = v2f{e0.x, o0.x}; *(v2f*)&brow[2]  = v2f{e0.y, o0.y};
            *(v2f*)&brow[4]  = v2f{e0.z, o0.z}; *(v2f*)&brow[6]  = v2f{e0.w, o0.w};
            *(v2f*)&brow[8]  = v2f{e1.x, o1.x}; *(v2f*)&brow[10] = v2f{e1.y, o1.y};
            *(v2f*)&brow[12] = v2f{e1.z, o1.z}; *(v2f*)&brow[14] = v2f{e1.w, o1.w};
        }
        __syncthreads();
    }

    // ---- store: D layout = VGPR r -> (M = r + 8*half, N = lr) ----
#pragma unroll
    for (int jt = 0; jt < 4; ++jt) {
        const int col = tileN + nB + jt * 16 + lr;
#pragma unroll
        for (int r = 0; r < 8; ++r) {
            const int row = rowBase0 + mB + r + 8 * half;
            Y[(size_t)row * NC + col] = acc[jt][r] + bias[row];
        }
    }
}

// ---------------------------------------------------------------------------
// Kernel 4: per-channel BN statistics (mean, rsqrt(var+eps)) over NC columns.
// ---------------------------------------------------------------------------
__global__ void __launch_bounds__(256)
bn_stats_kernel(const float* __restrict__ Y, float* __restrict__ mean,
                float* __restrict__ rstd, int NC) {
    __shared__ float ss[256], sq[256];
    const int o = blockIdx.x;
    float s = 0.f, q = 0.f;
    for (int j = threadIdx.x; j < NC; j += blockDim.x) {
        const float v = Y[(size_t)o * NC + j];
        s += v; q += v * v;
    }
    ss[threadIdx.x] = s; sq[threadIdx.x] = q;
    __syncthreads();
    for (int st = 128; st > 0; st >>= 1) {
        if (threadIdx.x < st) {
            ss[threadIdx.x] += ss[threadIdx.x + st];
            sq[threadIdx.x] += sq[threadIdx.x + st];
        }
        __syncthreads();
    }
    if (threadIdx.x == 0) {
        const float m = ss[0] / (float)NC;
        const float var = sq[0] / (float)NC - m * m;
        mean[o] = m;
        rstd[o] = rsqrtf(var + 1e-5f);
    }
}

// ---------------------------------------------------------------------------
// Kernel 5a: BN-normalize + ReLU in place ([M x NC] layout).
// ---------------------------------------------------------------------------
__global__ void __launch_bounds__(256)
bn_relu_inplace_kernel(float* __restrict__ Y, const float* __restrict__ mean,
                       const float* __restrict__ rstd, const float* __restrict__ g,
                       const float* __restrict__ beta, int NC) {
    const int i = blockIdx.x * blockDim.x + threadIdx.x;
    const int o = i / NC;
    const float v = (Y[i] - mean[o]) * rstd[o] * g[o] + beta[o];
    Y[i] = fmaxf(v, 0.f);
}

// ---------------------------------------------------------------------------
// Kernel 5b: final BN + ReLU, writing output in [B, C_out, N] layout.
// ---------------------------------------------------------------------------
__global__ void __launch_bounds__(256)
bn_relu_out_kernel(const float* __restrict__ Y, const float* __restrict__ mean,
                   const float* __restrict__ rstd, const float* __restrict__ g,
                   const float* __restrict__ beta, float* __restrict__ out, int NC) {
    const int i = blockIdx.x * blockDim.x + threadIdx.x;  // i = o*NC + j
    const int o = i / NC;
    const int j = i - o * NC;      // j = b*N + n
    const int b = j / N_;
    const int n = j - b * N_;
    const float v = (Y[i] - mean[o]) * rstd[o] * g[o] + beta[o];
    out[(size_t)(b * COUT_ + o) * N_ + n] = fmaxf(v, 0.f);
}

// ---------------------------------------------------------------------------
extern "C" void kernel_launch(void* const* d_in, const int* in_sizes, int n_in,
                              void* d_out, int out_size, void* d_ws, size_t ws_size,
                              hipStream_t stream) {
    (void)in_sizes; (void)n_in; (void)out_size; (void)ws_size;
    const float* pos1     = (const float*)d_in[0];
    const float* pos2     = (const float*)d_in[1];
    const float* feature1 = (const float*)d_in[2];
    const float* feature2 = (const float*)d_in[3];
    const float* W1    = (const float*)d_in[4];
    const float* b1    = (const float*)d_in[5];
    const float* g1    = (const float*)d_in[6];
    const float* beta1 = (const float*)d_in[7];
    const float* W2    = (const float*)d_in[8];
    const float* b2    = (const float*)d_in[9];
    const float* g2    = (const float*)d_in[10];
    const float* beta2 = (const float*)d_in[11];
    float* out = (float*)d_out;

    char* ws = (char*)d_ws;
    size_t off = 0;
    auto carve = [&](size_t bytes) -> void* {
        void* p = ws + off;
        off += (bytes + 255) & ~(size_t)255;
        return p;
    };
    int*   nnIdx = (int*)  carve(sizeof(int)   * (size_t)BN_ * 3);
    float* nnW   = (float*)carve(sizeof(float) * (size_t)BN_ * 3);
    float* feat  = (float*)carve(sizeof(float) * (size_t)CIN_  * BN_);
    float* y1    = (float*)carve(sizeof(float) * (size_t)CMID_ * BN_);
    float* y2    = (float*)carve(sizeof(float) * (size_t)COUT_ * BN_);
    float* mean1 = (float*)carve(sizeof(float) * CMID_);
    float* rstd1 = (float*)carve(sizeof(float) * CMID_);
    float* mean2 = (float*)carve(sizeof(float) * COUT_);
    float* rstd2 = (float*)carve(sizeof(float) * COUT_);

    three_nn_kernel<<<dim3(N_ / 256, B_), 256, 0, stream>>>(pos1, pos2, nnIdx, nnW);

    interp_concat_kernel<<<dim3(BN_ / 256, CIN_), 256, 0, stream>>>(
        feature1, feature2, nnIdx, nnW, feat);

    gemm_f32_wmma_kernel<<<dim3(BN_ / TN_, CMID_ / TM_), 256, 0, stream>>>(
        W1, b1, feat, y1, CMID_, CIN_, BN_);
    bn_stats_kernel<<<CMID_, 256, 0, stream>>>(y1, mean1, rstd1, BN_);
    bn_relu_inplace_kernel<<<(CMID_ * BN_) / 256, 256, 0, stream>>>(
        y1, mean1, rstd1, g1, beta1, BN_);

    gemm_f32_wmma_kernel<<<dim3(BN_ / TN_, COUT_ / TM_), 256, 0, stream>>>(
        W2, b2, y1, y2, COUT_, CMID_, BN_);
    bn_stats_kernel<<<COUT_, 256, 0, stream>>>(y2, mean2, rstd2, BN_);
    bn_relu_out_kernel<<<(COUT_ * BN_) / 256, 256, 0, stream>>>(
        y2, mean2, rstd2, g2, beta2, out, BN_);
}